// KalmanFilter_81870666596831
// MI455X (gfx1250) — compile-verified
//
#include <hip/hip_runtime.h>

// ---------------- Problem geometry ----------------
#define BATCH        64
#define CHAN         1024
#define W            512
#define ROWS_TOTAL   (BATCH * CHAN)          // 65536 independent rows

// ---------------- Kernel geometry -----------------
#define WAVE_SIZE        32
#define WAVES_PER_BLOCK  4
#define BLOCK_THREADS    (WAVES_PER_BLOCK * WAVE_SIZE)   // 128
#define ROWS_PER_WAVE    32                              // one row per lane
#define ROWS_PER_BLOCK   (WAVES_PER_BLOCK * ROWS_PER_WAVE) // 128
#define NBLOCKS          (ROWS_TOTAL / ROWS_PER_BLOCK)     // 512

#define CHUNK        32                      // columns per TDM tile
#define NCHUNK       (W / CHUNK)             // 16
#define LDS_STRIDE   (CHUNK + 1)             // 33 dwords: odd stride -> no bank conflicts

typedef __attribute__((ext_vector_type(4))) unsigned int v4u;
typedef __attribute__((ext_vector_type(8))) int          v8i;
typedef __attribute__((ext_vector_type(4))) int          v4i;

// ---------------------------------------------------------------------------
// TDM: DMA a [rows=32] x [cols=CHUNK] fp32 tile (row stride 512 elements in
// global) into LDS, padding 1 DWORD after every 32 DWORDs so the LDS tile has
// a 33-DWORD row stride (bank-conflict-free column reads).
// Descriptor layout per CDNA5 ISA ch.8 (CDNA5 ISA).
// This toolchain exposes the 6-arg builtin:
//   (uint32x4 g0, int32x8 g1, int32x4 g2, int32x4 g3, int32x8 g4, i32 cpol)
// ---------------------------------------------------------------------------
__device__ __forceinline__ void tdm_load_tile(unsigned lds_off,
                                              const float* gptr,
                                              unsigned rows_rem,
                                              unsigned cols_rem)
{
    unsigned long long ga = (unsigned long long)(uintptr_t)gptr;

    v4u g0;
    g0[0] = 1u;                                    // count=1, is_restore=0, gather=0
    g0[1] = lds_off;                               // lds_addr (bytes)
    g0[2] = (unsigned)(ga & 0xFFFFFFFFu);          // global_addr[31:0]
    g0[3] = (unsigned)((ga >> 32) & 0x01FFFFFFu)   // global_addr[56:32]
          | (2u << 30);                            // type = 2 ("image")

    v8i g1;
    // data_size=2 (4B) | pad_enable | pad_interval=4 (32 DW) | pad_amount=0 (1 DW)
    g1[0] = (int)((2u << 16) | (1u << 20) | (4u << 22));
    g1[1] = (int)((cols_rem & 0xFFFFu) << 16);                       // tensor_dim0[15:0]
    g1[2] = (int)(((cols_rem >> 16) & 0xFFFFu)                       // tensor_dim0[31:16]
          | ((rows_rem & 0xFFFFu) << 16));                           // tensor_dim1[15:0]
    g1[3] = (int)(((rows_rem >> 16) & 0xFFFFu)                       // tensor_dim1[31:16]
          | ((unsigned)CHUNK << 16));                                // tile_dim0
    g1[4] = (int)ROWS_PER_WAVE;                                      // tile_dim1=32, tile_dim2=0
    g1[5] = (int)W;                                                  // tensor_dim0_stride[31:0]
    g1[6] = 0;                                                       // stride hi / dim1_stride lo
    g1[7] = 0;

    v4i z4 = {0, 0, 0, 0};                         // groups 2/3: dims 2..4 unused
    v8i z8 = {0, 0, 0, 0, 0, 0, 0, 0};
    __builtin_amdgcn_tensor_load_to_lds(g0, g1, z4, z4, z8, 0);
}

// Kalman gain recurrence (depends only on scalars Q,R; shared by all rows).
// gK[0] = 1 so that x = 0; x += K*(z-x) reproduces x0 = z0 exactly.
__device__ __forceinline__ void compute_gains(float Q, float R, float* gK)
{
    float P = Q + R;
    gK[0] = 1.0f;
    for (int w = 1; w < W; ++w) {
        float Pp = P + Q;
        float K  = Pp / (Pp + R);
        P = (1.0f - K) * Pp;
        gK[w] = K;
    }
}

__global__ void kf_gains_kernel(const float* __restrict__ logQ,
                                const float* __restrict__ logR,
                                float* __restrict__ gK)
{
    if (threadIdx.x == 0 && blockIdx.x == 0)
        compute_gains(__expf(logQ[0]), __expf(logR[0]), gK);
}

// ---------------------------------------------------------------------------
// Main streaming scan: each wave owns 32 rows. Double-buffered TDM tiles in
// LDS; per-lane sequential Kalman update; float4 stores of the result.
// ---------------------------------------------------------------------------
__global__ __launch_bounds__(BLOCK_THREADS)
void kf_scan_kernel(const float* __restrict__ z,
                    const float* __restrict__ gK,     // precomputed gains (or null)
                    const float* __restrict__ logQ,
                    const float* __restrict__ logR,
                    float* __restrict__ out,
                    int use_gains)
{
    __shared__ float sK[W];
    __shared__ float tile[WAVES_PER_BLOCK][2][ROWS_PER_WAVE * LDS_STRIDE];

    if (use_gains) {
        for (int i = threadIdx.x; i < W; i += BLOCK_THREADS) sK[i] = gK[i];
    } else if (threadIdx.x == 0) {
        compute_gains(__expf(logQ[0]), __expf(logR[0]), sK);
    }
    __syncthreads();

    const int wave    = threadIdx.x >> 5;
    const int lane    = threadIdx.x & (WAVE_SIZE - 1);
    const int rowBase = (int)(blockIdx.x * WAVES_PER_BLOCK + wave) * ROWS_PER_WAVE;

    const float* gbase = z + (size_t)rowBase * W;
    float*       obase = out + (size_t)(rowBase + lane) * W;

    const unsigned ldsA = (unsigned)(uintptr_t)&tile[wave][0][0];
    const unsigned ldsB = (unsigned)(uintptr_t)&tile[wave][1][0];
    const unsigned rows_rem = (unsigned)(ROWS_TOTAL - rowBase);

    // Prime the pipeline: chunk 0 into buffer A.
    tdm_load_tile(ldsA, gbase, rows_rem, (unsigned)W);

    float x = 0.0f;                      // with sK[0]==1 -> x0 = z0 exactly

#pragma unroll
    for (int c = 0; c < NCHUNK; ++c) {
        if (c + 1 < NCHUNK) {
            // Issue the next chunk's DMA, then wait until chunk c landed
            // (exactly one younger DMA outstanding). Immediates only: the
            // s_wait_tensorcnt builtin demands a constant argument.
            tdm_load_tile((c & 1) ? ldsA : ldsB,
                          gbase + (size_t)(c + 1) * CHUNK,
                          rows_rem, (unsigned)(W - (c + 1) * CHUNK));
            __builtin_amdgcn_s_wait_tensorcnt(1);
        } else {
            __builtin_amdgcn_s_wait_tensorcnt(0);
        }
        asm volatile("" ::: "memory");   // DMA wrote LDS behind the compiler's back

        const float* tb = &tile[wave][c & 1][lane * LDS_STRIDE];
        float4* op = (float4*)(obase + c * CHUNK);

#pragma unroll
        for (int j = 0; j < CHUNK / 4; ++j) {
            float4 o;
            x = fmaf(sK[c * CHUNK + 4 * j + 0], tb[4 * j + 0] - x, x); o.x = x;
            x = fmaf(sK[c * CHUNK + 4 * j + 1], tb[4 * j + 1] - x, x); o.y = x;
            x = fmaf(sK[c * CHUNK + 4 * j + 2], tb[4 * j + 2] - x, x); o.z = x;
            x = fmaf(sK[c * CHUNK + 4 * j + 3], tb[4 * j + 3] - x, x); o.w = x;
            op[j] = o;
        }
    }
}

extern "C" void kernel_launch(void* const* d_in, const int* in_sizes, int n_in,
                              void* d_out, int out_size, void* d_ws, size_t ws_size,
                              hipStream_t stream)
{
    (void)in_sizes; (void)n_in; (void)out_size;

    const float* z    = (const float*)d_in[0];
    const float* logQ = (const float*)d_in[1];
    const float* logR = (const float*)d_in[2];
    float*       out  = (float*)d_out;

    const int use_ws = (d_ws != nullptr && ws_size >= W * sizeof(float)) ? 1 : 0;
    float* gK = (float*)d_ws;

    if (use_ws)
        kf_gains_kernel<<<1, 1, 0, stream>>>(logQ, logR, gK);

    kf_scan_kernel<<<dim3(NBLOCKS), dim3(BLOCK_THREADS), 0, stream>>>(
        z, gK, logQ, logR, out, use_ws);
}